// PopPredict_87823491269059
// MI455X (gfx1250) — compile-verified
//
#include <hip/hip_runtime.h>
#include <math.h>

#define BATCH   4096
#define TLEN    2049
#define EDIM    128
#define BN_EPS  1e-5f
// log2(0.9)
#define L2_09  (-0.15200309344504997f)

typedef __attribute__((ext_vector_type(2))) float v2f;
typedef __attribute__((ext_vector_type(8))) float v8f;

// ---------------------------------------------------------------------------
// Kernel 0: fold BN + FC into combined per-feature weights.
// wcomb layout (5 x 128):
//   [0:128)   item   weights  = a_t[128+k]
//   [128:256) time_e weights  = a_t[256+k] - a_t[k]      (gap fold: -gap part)
//   [256:384) rel_e  weights  = a_t[k]     + a_t[384+k]  (gap fold: +gap part)
//   [384:512) cat    weights  = a_s[k]
//   [512:640) store  weights  = a_s[128+k]
// consts[0] = c_t, consts[1] = c_s
// ---------------------------------------------------------------------------
__global__ __launch_bounds__(256) void pp_prep_kernel(
    const float* __restrict__ bn_t_gamma, const float* __restrict__ bn_t_beta,
    const float* __restrict__ bn_t_mean,  const float* __restrict__ bn_t_var,
    const float* __restrict__ fc_t_w,     const float* __restrict__ fc_t_b,
    const float* __restrict__ bn_s_gamma, const float* __restrict__ bn_s_beta,
    const float* __restrict__ bn_s_mean,  const float* __restrict__ bn_s_var,
    const float* __restrict__ fc_s_w,     const float* __restrict__ fc_s_b,
    float* __restrict__ wcomb, float* __restrict__ consts)
{
    __shared__ float a_t[512];
    __shared__ float a_s[256];
    const int t = threadIdx.x;
    for (int k = t; k < 512; k += 256)
        a_t[k] = rsqrtf(bn_t_var[k] + BN_EPS) * bn_t_gamma[k] * fc_t_w[k];
    a_s[t] = rsqrtf(bn_s_var[t] + BN_EPS) * bn_s_gamma[t] * fc_s_w[t];
    __syncthreads();
    if (t < 128) {
        const int k = t;
        wcomb[k]       = a_t[128 + k];
        wcomb[128 + k] = a_t[256 + k] - a_t[k];
        wcomb[256 + k] = a_t[k] + a_t[384 + k];
        wcomb[384 + k] = a_s[k];
        wcomb[512 + k] = a_s[128 + k];
    }
    if (t == 0) {
        float ct = fc_t_b[0];
        for (int k = 0; k < 512; ++k)
            ct += (bn_t_beta[k] - bn_t_mean[k] * rsqrtf(bn_t_var[k] + BN_EPS) * bn_t_gamma[k]) * fc_t_w[k];
        consts[0] = ct;
        float cs = fc_s_b[0];
        for (int k = 0; k < 256; ++k)
            cs += (bn_s_beta[k] - bn_s_mean[k] * rsqrtf(bn_s_var[k] + BN_EPS) * bn_s_gamma[k]) * fc_s_w[k];
        consts[1] = cs;
    }
}

// ---------------------------------------------------------------------------
// Kernel 1: EMA evaluated only at idx = clip(time-1, 0) via closed form:
//   c_idx = 0.9^idx * x0 + 0.1 * sum_{j=1..idx} 0.9^(idx-j) * x_j
// One wave per batch row; lanes stride the time axis, shuffle-reduce.
// ---------------------------------------------------------------------------
__global__ __launch_bounds__(256) void pp_ema_kernel(
    const float* __restrict__ pop_history, const int* __restrict__ time_idx,
    float* __restrict__ pop_out)
{
    const int wave = threadIdx.x >> 5;
    const int lane = threadIdx.x & 31;
    const int b = blockIdx.x * 8 + wave;
    if (b >= BATCH) return;

    int idx = time_idx[b] - 1;
    if (idx < 0) idx = 0;
    const float* __restrict__ row = pop_history + (size_t)b * TLEN;

    float sum = 0.0f;
    for (int j = lane; j <= idx; j += 32) {
        __builtin_prefetch(row + j + 512, 0, 1);   // global_prefetch_b8
        const float w = __builtin_exp2f((float)(idx - j) * L2_09) *
                        ((j == 0) ? 1.0f : 0.1f);
        sum = fmaf(w, row[j], sum);
    }
    for (int s = 16; s > 0; s >>= 1)
        sum += __shfl_xor(sum, s, 32);
    if (lane == 0) pop_out[b] = sum;
}

// ---------------------------------------------------------------------------
// Kernel 2: embedding gathers + 5 folded dot products per row, computed with
// V_WMMA_F32_16X16X4_F32. One wave handles 16 batch rows (a tile).
//   A (16x4 f32): rows = batch rows, K = 4 consecutive features.
//     lane L<16: M=L, VGPR0=K0, VGPR1=K1 ; lane L+16: M=L, VGPR0=K2, VGPR1=K3
//   B (4x16 f32): column j carries weight-vector j only while the K-loop is
//     inside segment j (branchless lane mask) -> D column j = dot product j.
//   D (16x16 f32, v8f): acc[i] lane n<16 = D[i][n], lane 16+n = D[i+8][n].
// The 5 segments are fully specialized (no pointer array) so every embedding
// stream stays on the GLOBAL path (global_load_b64, LOADcnt only) instead of
// decaying to flat_load (which also burns DScnt / the LDS pipe).
// ---------------------------------------------------------------------------
__global__ __launch_bounds__(256) void pp_embed_wmma_kernel(
    const int* __restrict__ item, const int* __restrict__ time_idx,
    const int* __restrict__ release_time, const int* __restrict__ category,
    const int* __restrict__ store,
    const float* __restrict__ item_emb, const float* __restrict__ cat_emb,
    const float* __restrict__ store_emb, const float* __restrict__ time_emb,
    const float* __restrict__ wcomb, const float* __restrict__ consts,
    float* __restrict__ time_raw, float* __restrict__ side_raw)
{
    const int wave = threadIdx.x >> 5;
    const int lane = threadIdx.x & 31;
    const int tile = blockIdx.x * 8 + wave;     // 16 rows per tile
    const int m    = lane & 15;
    const int row  = tile * 16 + m;
    const int off  = (lane < 16) ? 0 : 2;

    v8f acc = {};

    // One segment = one (embedding row, weight vector, output column).
    auto segment = [&](const float* __restrict__ erow, int j) {
        const float act = (m == j) ? 1.0f : 0.0f;   // column mask (branchless)
        const float* __restrict__ ep = erow + off;
        const float* __restrict__ wp = wcomb + j * EDIM + off;
        #pragma unroll 8
        for (int kl = 0; kl < EDIM; kl += 4) {
            const v2f a  = *(const v2f*)(ep + kl);
            const v2f bw = *(const v2f*)(wp + kl);
            v2f b;
            b.x = bw.x * act;
            b.y = bw.y * act;
            acc = __builtin_amdgcn_wmma_f32_16x16x4_f32(
                false, a, false, b, (short)0, acc, false, false);
        }
    };

    segment(item_emb  + (size_t)item[row]         * EDIM, 0);
    segment(time_emb  + (size_t)time_idx[row]     * EDIM, 1);
    segment(time_emb  + (size_t)release_time[row] * EDIM, 2);
    segment(cat_emb   + (size_t)category[row]     * EDIM, 3);
    segment(store_emb + (size_t)store[row]        * EDIM, 4);

    const float c_t = consts[0];
    const float c_s = consts[1];
    const int shbase = lane & 16;   // lanes <16 read cols 0..4, >=16 read 16..20

    #pragma unroll
    for (int i = 0; i < 8; ++i) {
        const float d_item  = __shfl(acc[i], shbase + 0, 32);
        const float d_time  = __shfl(acc[i], shbase + 1, 32);
        const float d_rel   = __shfl(acc[i], shbase + 2, 32);
        const float d_cat   = __shfl(acc[i], shbase + 3, 32);
        const float d_store = __shfl(acc[i], shbase + 4, 32);
        if (m == 0) {
            const int r = tile * 16 + i + ((lane >> 4) << 3);
            time_raw[r] = fmaxf(d_item + d_time + d_rel + c_t, 0.0f);
            side_raw[r] = d_cat + d_store + c_s;
        }
    }
}

// ---------------------------------------------------------------------------
// Kernel 3: softmax over the 3 attention weights + weighted combine.
// d_out = [w_pop(B) | w_time(B) | w_side(B) | total(B)]
// ---------------------------------------------------------------------------
__global__ __launch_bounds__(256) void pp_combine_kernel(
    const float* __restrict__ attn_w, const float* __restrict__ pop_out,
    const float* __restrict__ time_raw, const float* __restrict__ side_raw,
    float* __restrict__ out)
{
    const int b = blockIdx.x * 256 + threadIdx.x;
    if (b >= BATCH) return;
    const float a0 = attn_w[0], a1 = attn_w[1], a2 = attn_w[2];
    const float mx = fmaxf(a0, fmaxf(a1, a2));
    const float e0 = __expf(a0 - mx), e1 = __expf(a1 - mx), e2 = __expf(a2 - mx);
    const float inv = 1.0f / (e0 + e1 + e2);
    const float wp = pop_out[b]  * (e0 * inv);
    const float wt = time_raw[b] * (e1 * inv);
    const float ws = side_raw[b] * (e2 * inv);
    out[b]             = wp;
    out[BATCH + b]     = wt;
    out[2 * BATCH + b] = ws;
    out[3 * BATCH + b] = wp + wt + ws;
}

// ---------------------------------------------------------------------------
extern "C" void kernel_launch(void* const* d_in, const int* in_sizes, int n_in,
                              void* d_out, int out_size, void* d_ws, size_t ws_size,
                              hipStream_t stream)
{
    const int*   item         = (const int*)  d_in[0];
    const int*   time_idx     = (const int*)  d_in[1];
    const int*   release_time = (const int*)  d_in[2];
    const int*   category     = (const int*)  d_in[3];
    const int*   store        = (const int*)  d_in[4];
    const float* pop_history  = (const float*)d_in[5];
    const float* item_emb     = (const float*)d_in[6];
    const float* cat_emb      = (const float*)d_in[7];
    const float* store_emb    = (const float*)d_in[8];
    const float* time_emb     = (const float*)d_in[9];
    const float* attn_w       = (const float*)d_in[10];
    const float* bn_t_gamma   = (const float*)d_in[11];
    const float* bn_t_beta    = (const float*)d_in[12];
    const float* bn_t_mean    = (const float*)d_in[13];
    const float* bn_t_var     = (const float*)d_in[14];
    const float* fc_t_w       = (const float*)d_in[15];
    const float* fc_t_b       = (const float*)d_in[16];
    const float* bn_s_gamma   = (const float*)d_in[17];
    const float* bn_s_beta    = (const float*)d_in[18];
    const float* bn_s_mean    = (const float*)d_in[19];
    const float* bn_s_var     = (const float*)d_in[20];
    const float* fc_s_w       = (const float*)d_in[21];
    const float* fc_s_b       = (const float*)d_in[22];

    float* ws       = (float*)d_ws;
    float* wcomb    = ws;               // 640
    float* consts   = ws + 640;         // 2
    float* pop_out  = ws + 642;         // BATCH
    float* time_raw = pop_out + BATCH;  // BATCH
    float* side_raw = time_raw + BATCH; // BATCH

    pp_prep_kernel<<<1, 256, 0, stream>>>(
        bn_t_gamma, bn_t_beta, bn_t_mean, bn_t_var, fc_t_w, fc_t_b,
        bn_s_gamma, bn_s_beta, bn_s_mean, bn_s_var, fc_s_w, fc_s_b,
        wcomb, consts);

    pp_ema_kernel<<<BATCH / 8, 256, 0, stream>>>(pop_history, time_idx, pop_out);

    pp_embed_wmma_kernel<<<BATCH / (16 * 8), 256, 0, stream>>>(
        item, time_idx, release_time, category, store,
        item_emb, cat_emb, store_emb, time_emb,
        wcomb, consts, time_raw, side_raw);

    pp_combine_kernel<<<(BATCH + 255) / 256, 256, 0, stream>>>(
        attn_w, pop_out, time_raw, side_raw, (float*)d_out);
}